// DEINA_48979807044193
// MI455X (gfx1250) — compile-verified
//
#include <hip/hip_runtime.h>

typedef float v2f __attribute__((ext_vector_type(2)));
typedef float v4f __attribute__((ext_vector_type(4)));
typedef float v8f __attribute__((ext_vector_type(8)));

__device__ __forceinline__ v8f wmma_f32(v2f a, v2f b, v8f c) {
  // D = A(16x4) * B(4x16) + C(16x16), fp32, wave32
  return __builtin_amdgcn_wmma_f32_16x16x4_f32(false, a, false, b, (short)0, c,
                                               false, false);
}

#define HPAD 260   // 256-col rows padded: stride%64==4 -> conflict-free frags
#define GPAD 84    // 80-col staging rows
#define RPAD 68    // 64-col partial-reduction rows (stride%64==4)
#define KPAD 84
#define BWPAD 20
#define TPB 256            // tokens per block (MLP)
#define NT (TPB / 16)      // token tiles per block

// ---------------------------------------------------------------------------
// Kernel 1: weight-stationary 3-layer MLP, 16 symmetric waves (512 threads).
// Wave w keeps in registers:
//   - W1 rows [16w, 16w+16) as 64 f32 WMMA A-frags (128 VGPRs)
//   - W0 rows [16w, 16w+16) (8 VGPRs) + b0/b1 slices
//   - W2 tile nt=w&3, k-quarter kq=w>>2 as 16 A-frags (32 VGPRs)
// Per 16-token tile:  L0 -> S1 -> (drain prev | L1) -> S2 -> L2 partials
//   -> S3 -> kq==0 waves reduce partials -> gb.  84 WMMA per wave per tile
// (= ideal 84 wmma/token, perfectly balanced).  W1/W2 are read from global
// once per 256 tokens (~330 MB total vs 5.2 GB for the naive scheme).
// ---------------------------------------------------------------------------
__global__ __launch_bounds__(512, 1)
void fused_mlp_kernel(const float* __restrict__ xs,
                      const float* __restrict__ W0, const float* __restrict__ b0v,
                      const float* __restrict__ W1, const float* __restrict__ b1v,
                      const float* __restrict__ W2,
                      float* __restrict__ y2s,     // (2048,127,80)
                      float* __restrict__ y0_ws)   // (2048,80)
{
  __shared__ __align__(16) float h1[16 * HPAD];
  __shared__ __align__(16) float h2[16 * HPAD];
  __shared__ __align__(16) float red[3 * 16 * RPAD];
  __shared__ __align__(16) float gb[16 * GPAD];

  const int lane  = threadIdx.x & 31;
  const int wave  = threadIdx.x >> 5;      // 0..15
  const int lrow  = lane & 15;
  const int khalf = (lane >> 4) << 1;      // K sub-offset 0 / 2
  const int mhalf = (lane >> 4) << 3;      // C row offset 0 / 8
  const int tok0  = blockIdx.x * TPB;
  const int nt    = wave & 3;              // W2 output tile
  const int kq    = wave >> 2;             // W2 k-quarter

  // ---- stationary weights ----
  v2f wA[64];                              // W1 slab
  {
    const float* w1r = W1 + (wave * 16 + lrow) * 256;
#pragma unroll
    for (int kk = 0; kk < 64; ++kk)
      wA[kk] = *(const v2f*)(w1r + kk * 4 + khalf);
  }
  v2f wA2[16];                             // W2 quarter slab
  {
    const float* w2r = W2 + (nt * 16 + lrow) * 256 + kq * 64;
#pragma unroll
    for (int j = 0; j < 16; ++j)
      wA2[j] = *(const v2f*)(w2r + j * 4 + khalf);
  }
  v2f wA0[4];
  float bs0[8], bs1[8];
  {
    const int nb = wave * 16;
    const float* w0r = W0 + (nb + lrow) * 16;
#pragma unroll
    for (int kk = 0; kk < 4; ++kk)
      wA0[kk] = *(const v2f*)(w0r + kk * 4 + khalf);
#pragma unroll
    for (int v = 0; v < 8; ++v) {
      bs0[v] = b0v[nb + mhalf + v];
      bs1[v] = b1v[nb + mhalf + v];
    }
  }

  for (int i = 0; i <= NT; ++i) {
    // ---- a) layer 0 on tile i: C[feature][token] ----
    if (i < NT) {
      const float* xr = xs + (tok0 + i * 16 + lrow) * 16;
      v8f a0 = {}, a1 = {};
      a0 = wmma_f32(wA0[0], *(const v2f*)(xr + 0 + khalf), a0);
      a1 = wmma_f32(wA0[1], *(const v2f*)(xr + 4 + khalf), a1);
      a0 = wmma_f32(wA0[2], *(const v2f*)(xr + 8 + khalf), a0);
      a1 = wmma_f32(wA0[3], *(const v2f*)(xr + 12 + khalf), a1);
      v8f acc = a0 + a1;
      v4f lo, hi;
#pragma unroll
      for (int v = 0; v < 4; ++v) {
        float x0 = acc[v] + bs0[v];         lo[v] = x0 > 0.f ? x0 : 0.f;
        float x1 = acc[v + 4] + bs0[v + 4]; hi[v] = x1 > 0.f ? x1 : 0.f;
      }
      float* out = &h1[lrow * HPAD + wave * 16 + mhalf];
      *(v4f*)(out)     = lo;     // lane owns 8 consecutive features
      *(v4f*)(out + 4) = hi;
    }
    __syncthreads();   // S1: h1 ready; gb from iter i-1 safe to drain

    // ---- b) drain tile i-1 (coalesced float2 rows; x cols from xs) ----
    if (i >= 1) {
      const int btok = tok0 + (i - 1) * 16;
#pragma unroll
      for (int r = 0; r < 2; ++r) {
        const int j = r * 512 + (int)threadIdx.x;   // 0..639 valid
        if (j < 640) {
          const int tk  = j / 40;
          const int cc  = (j % 40) * 2;
          const int tok = btok + tk;
          const int bb  = tok >> 7, tt = tok & 127;
          v2f val;
          if (cc < 16) val = *(const v2f*)(&xs[tok * 16 + cc]);
          else         val = *(const v2f*)(&gb[tk * GPAD + cc]);
          if (tt >= 1)
            __builtin_nontemporal_store(val,
                (v2f*)&y2s[(bb * 127 + tt - 1) * 80 + cc]);
          else
            *(v2f*)(&y0_ws[bb * 80 + cc]) = val;
        }
      }
    }

    // ---- c) layer 1 on tile i: K=256, dual accumulation chains ----
    if (i < NT) {
      const float* hb = &h1[lrow * HPAD + khalf];
      v8f a0 = {}, a1 = {};
#pragma unroll
      for (int kk = 0; kk < 64; kk += 2) {
        a0 = wmma_f32(wA[kk],     *(const v2f*)(hb + kk * 4),       a0);
        a1 = wmma_f32(wA[kk + 1], *(const v2f*)(hb + (kk + 1) * 4), a1);
      }
      v8f acc = a0 + a1;
      v4f lo, hi;
#pragma unroll
      for (int v = 0; v < 4; ++v) {
        float x0 = acc[v] + bs1[v];         lo[v] = x0 > 0.f ? x0 : 0.f;
        float x1 = acc[v + 4] + bs1[v + 4]; hi[v] = x1 > 0.f ? x1 : 0.f;
      }
      float* out = &h2[lrow * HPAD + wave * 16 + mhalf];
      *(v4f*)(out)     = lo;
      *(v4f*)(out + 4) = hi;
    }
    __syncthreads();   // S2: h2 ready

    // ---- d) layer 2 partial: k-range [64*kq, 64*kq+64) of output tile nt ----
    v8f gacc = {};
    if (i < NT) {
      const float* hb2 = &h2[lrow * HPAD + kq * 64 + khalf];
      v8f g0 = {}, g1 = {};
#pragma unroll
      for (int j = 0; j < 16; j += 2) {
        g0 = wmma_f32(wA2[j],     *(const v2f*)(hb2 + j * 4),       g0);
        g1 = wmma_f32(wA2[j + 1], *(const v2f*)(hb2 + (j + 1) * 4), g1);
      }
      gacc = g0 + g1;
      if (kq > 0) {
        float* rp = &red[((kq - 1) * 16 + lrow) * RPAD + nt * 16 + mhalf];
        v4f lo, hi;
#pragma unroll
        for (int v = 0; v < 4; ++v) { lo[v] = gacc[v]; hi[v] = gacc[v + 4]; }
        *(v4f*)(rp)     = lo;
        *(v4f*)(rp + 4) = hi;
      }
    }
    __syncthreads();   // S3: partials ready

    // ---- e) kq==0 waves: reduce 4 partials -> gb cols 16..79 ----
    if (i < NT && kq == 0) {
      const int off = lrow * RPAD + nt * 16 + mhalf;
      v4f lo, hi;
#pragma unroll
      for (int v = 0; v < 4; ++v) { lo[v] = gacc[v]; hi[v] = gacc[v + 4]; }
#pragma unroll
      for (int r = 0; r < 3; ++r) {
        lo += *(const v4f*)(&red[r * 16 * RPAD + off]);
        hi += *(const v4f*)(&red[r * 16 * RPAD + off + 4]);
      }
      float* out = &gb[lrow * GPAD + 16 + nt * 16 + mhalf];
      *(v4f*)(out)     = lo;
      *(v4f*)(out + 4) = hi;
    }
  }
}

// ---------------------------------------------------------------------------
// Kernel 2: recurrent scan  y_{t+1} = y_t @ K^T + us[:,t,:] @ Bw^T
// One block per 16 batch rows; 5 waves = 5 N-tiles of the 80-wide state.
// Ping-pong LDS state -> single barrier per step; dual accumulation chains.
// ---------------------------------------------------------------------------
__global__ __launch_bounds__(160, 1)
void scan_kernel(const float* __restrict__ us,      // (2048,128,16)
                 const float* __restrict__ Bw,      // (80,16)
                 const float* __restrict__ Km,      // (80,80)
                 const float* __restrict__ y0_ws,   // (2048,80)
                 float* __restrict__ y2p)           // (2048,127,80)
{
  __shared__ __align__(16) float Kl[80 * KPAD];
  __shared__ __align__(16) float Bl[80 * BWPAD];
  __shared__ __align__(16) float yb[2][16 * KPAD];

  const int lane  = threadIdx.x & 31;
  const int wave  = threadIdx.x >> 5;     // 0..4 : N-tile
  const int lrow  = lane & 15;
  const int khalf = (lane >> 4) << 1;
  const int mhalf = (lane >> 4) << 3;
  const int b0    = blockIdx.x * 16;

  for (int i = threadIdx.x; i < 80 * 80; i += 160)
    Kl[(i / 80) * KPAD + (i % 80)] = Km[i];
  for (int i = threadIdx.x; i < 80 * 16; i += 160)
    Bl[(i / 16) * BWPAD + (i % 16)] = Bw[i];
  for (int i = threadIdx.x; i < 16 * 80; i += 160)
    yb[0][(i / 80) * KPAD + (i % 80)] = y0_ws[b0 * 80 + i];
  __syncthreads();

  const int ncol = wave * 16 + lrow;
  const float* Bn = &Bl[ncol * BWPAD + khalf];   // B[k][n] = Bw[n][k]
  const float* Kn = &Kl[ncol * KPAD + khalf];    // B[k][n] = K[n][k]

  int p = 0;
  for (int t = 0; t < 127; ++t) {
    v8f a0 = {}, a1 = {};
    // Bu contribution: A = us[b0..b0+15, t, :] (16x16)
    const float* urow = us + ((b0 + lrow) * 128 + t) * 16 + khalf;
    a0 = wmma_f32(*(const v2f*)(urow + 0), *(const v2f*)(Bn + 0), a0);
    a1 = wmma_f32(*(const v2f*)(urow + 4), *(const v2f*)(Bn + 4), a1);
    a0 = wmma_f32(*(const v2f*)(urow + 8), *(const v2f*)(Bn + 8), a0);
    a1 = wmma_f32(*(const v2f*)(urow + 12), *(const v2f*)(Bn + 12), a1);
    // y @ K^T contribution: A = y_t tile from LDS (16x80)
    const float* yin = &yb[p][lrow * KPAD + khalf];
#pragma unroll
    for (int kk = 0; kk < 20; kk += 2) {
      a0 = wmma_f32(*(const v2f*)(yin + kk * 4),       *(const v2f*)(Kn + kk * 4),       a0);
      a1 = wmma_f32(*(const v2f*)(yin + (kk + 1) * 4), *(const v2f*)(Kn + (kk + 1) * 4), a1);
    }
    v8f acc = a0 + a1;
    // write next state + streamed output
    float* yout = &yb[1 - p][0];
#pragma unroll
    for (int v = 0; v < 8; ++v) {
      const int m = v + mhalf;
      const float val = acc[v];
      yout[m * KPAD + ncol] = val;
      __builtin_nontemporal_store(val, &y2p[((b0 + m) * 127 + t) * 80 + ncol]);
    }
    p ^= 1;
    __syncthreads();
  }
}

// ---------------------------------------------------------------------------
extern "C" void kernel_launch(void* const* d_in, const int* in_sizes, int n_in,
                              void* d_out, int out_size, void* d_ws, size_t ws_size,
                              hipStream_t stream) {
  (void)in_sizes; (void)n_in; (void)out_size; (void)ws_size;
  const float* xs = (const float*)d_in[0];
  const float* us = (const float*)d_in[1];
  const float* W0 = (const float*)d_in[2];
  const float* b0 = (const float*)d_in[3];
  const float* W1 = (const float*)d_in[4];
  const float* b1 = (const float*)d_in[5];
  const float* W2 = (const float*)d_in[6];
  const float* Bw = (const float*)d_in[7];
  const float* Km = (const float*)d_in[8];

  float* y2s = (float*)d_out;                      // output 0: (2048,127,80)
  float* y2p = y2s + (size_t)2048 * 127 * 80;      // output 1: (2048,127,80)
  float* y0  = (float*)d_ws;                       // scratch:  (2048,80)

  fused_mlp_kernel<<<(2048 * 128) / TPB, 512, 0, stream>>>(xs, W0, b0, W1, b1,
                                                           W2, y2s, y0);
  scan_kernel<<<2048 / 16, 160, 0, stream>>>(us, Bw, Km, y0, y2p);
}